// QuanvolutionAdvanced_65481071398647
// MI455X (gfx1250) — compile-verified
//
#include <hip/hip_runtime.h>
#include <math.h>

// ---------------------------------------------------------------------------
// QuanvolutionAdvanced on MI455X (gfx1250, wave32).
//
// Restructure: the 2 parameterized layers form a data-independent 16x16
// complex unitary U(params); the RY-encoded state is purely real. The
// dominant work is a (B*196 x 16) x (16 x 32) f32 GEMM -> V_WMMA_F32_16X16X4_F32.
// ---------------------------------------------------------------------------

typedef __attribute__((ext_vector_type(2))) float v2f;
typedef __attribute__((ext_vector_type(8))) float v8f;

#define BATCH   4096
#define NPATCH  196
#define NTILES  13            // ceil(196/16)
#define TPB_B   (NTILES * 32) // 416 threads = 13 waves, one WMMA tile each

// workspace layout (in floats):
//   Br[16*16] : Br[k*16+n] = Re(U[n][k])   (WMMA B-matrix row order)
//   Bi[16*16] : Bi[k*16+n] = Im(U[n][k])
//   ctx[BATCH*4] : per-sample mean PauliZ expectations
#define WS_BR  0
#define WS_BI  256
#define WS_CTX 512

// ---------------------------------------------------------------------------
// Kernel A: build the fixed 16x16 complex unitary, one column per thread.
// All register-array indices are unroll-constants (no scratch).
// ---------------------------------------------------------------------------
#define APPLY_CNOT(cm, tm)                                            \
  {                                                                   \
    _Pragma("unroll")                                                 \
    for (int idx = 0; idx < 16; ++idx) {                              \
      if ((idx & (cm)) && !(idx & (tm))) {                            \
        float2 t = v[idx];                                            \
        v[idx] = v[idx | (tm)];                                       \
        v[idx | (tm)] = t;                                            \
      }                                                               \
    }                                                                 \
  }

__global__ __launch_bounds__(16) void build_unitary(
    const float* __restrict__ params, float* __restrict__ ws) {
  const int col = threadIdx.x;  // column k of U (circuit applied to e_k)
  float2 v[16];
#pragma unroll
  for (int i = 0; i < 16; ++i) v[i] = make_float2((i == col) ? 1.f : 0.f, 0.f);

#pragma unroll
  for (int l = 0; l < 2; ++l) {
#pragma unroll
    for (int w = 0; w < 4; ++w) {
      const int mask = 8 >> w;  // wire w <-> bit (3-w)
      // RZ(params[l][w][0]):  b=0 -> e^{-i hz},  b=1 -> e^{+i hz}
      float hz = 0.5f * params[(l * 4 + w) * 2 + 0];
      float cz = cosf(hz), sz = sinf(hz);
#pragma unroll
      for (int idx = 0; idx < 16; ++idx) {
        float sg = (idx & mask) ? sz : -sz;  // phase sin with sign
        float2 a = v[idx];
        v[idx] = make_float2(a.x * cz - a.y * sg, a.x * sg + a.y * cz);
      }
      // RX(params[l][w][1]): [[cx, -i sx], [-i sx, cx]]
      float hx = 0.5f * params[(l * 4 + w) * 2 + 1];
      float cx = cosf(hx), sx = sinf(hx);
#pragma unroll
      for (int idx = 0; idx < 16; ++idx) {
        if (idx & mask) continue;
        const int j = idx | mask;
        float2 a = v[idx], b = v[j];
        v[idx] = make_float2(cx * a.x + sx * b.y, cx * a.y - sx * b.x);
        v[j]   = make_float2(sx * a.y + cx * b.x, -sx * a.x + cx * b.y);
      }
    }
    // CNOT chain (0,1),(1,2),(2,3),(3,0); wire w mask = 8>>w
    APPLY_CNOT(8, 4);
    APPLY_CNOT(4, 2);
    APPLY_CNOT(2, 1);
    APPLY_CNOT(1, 8);
  }

  float* Br = ws + WS_BR;
  float* Bi = ws + WS_BI;
#pragma unroll
  for (int n = 0; n < 16; ++n) {  // column col -> U[n][col]
    Br[col * 16 + n] = v[n].x;
    Bi[col * 16 + n] = v[n].y;
  }
}

// ---------------------------------------------------------------------------
// Kernel B: per batch sample, 13 waves; each wave = one 16-patch WMMA tile.
// A (16x16 real states) x B (16x32 = [Ur^T | Ui^T]) via 8x v_wmma_f32_16x16x4.
// ---------------------------------------------------------------------------
__global__ __launch_bounds__(TPB_B) void quanv_encode(
    const float* __restrict__ x, float* __restrict__ ws) {
  __shared__ float eacc[NTILES * 2][4];  // per half-wave partial exp sums

  const int bb   = blockIdx.x;
  const int tid  = threadIdx.x;
  const int tile = tid >> 5;
  const int lane = tid & 31;
  const int h    = lane >> 4;  // half-wave: selects K pair (A/B frags)
  const int n    = lane & 15;  // A-frag row M; D-frag column N (state index)

  // ---- encode: real state amplitudes for patch p = tile*16 + M (M = n) ----
  const int p      = tile * 16 + n;
  const bool valid = (p < NPATCH);
  const int pc     = valid ? p : 0;
  const int pr     = pc / 14, pcc = pc % 14;
  const float* xb  = x + (size_t)bb * 784;
  const int base   = (2 * pr) * 28 + 2 * pcc;
  const float a0 = xb[base], a1 = xb[base + 1];
  const float a2 = xb[base + 28], a3 = xb[base + 29];

  float c0, s0, c1, s1, c2, s2, c3, s3;
  __sincosf(0.5f * a0, &s0, &c0);
  __sincosf(0.5f * a1, &s1, &c1);
  __sincosf(0.5f * a2, &s2, &c2);
  __sincosf(0.5f * a3, &s3, &c3);

  // amplitude(idx = b0b1b2b3) = pA[b0b1] * pB[b2b3]; invalid patches -> 0
  const float z = valid ? 1.0f : 0.0f;
  const float pA0 = z * c0 * c1, pA1 = z * c0 * s1;
  const float pA2 = z * s0 * c1, pA3 = z * s0 * s1;
  const float pB0 = c2 * c3, pB1 = c2 * s3, pB2 = s2 * c3, pB3 = s2 * s3;
  // This lane's K pair is k0 = 4*kk + 2h: amp[4kk+2h+j] = pA[kk] * pB[2h+j].
  const float qx = h ? pB2 : pB0;
  const float qy = h ? pB3 : pB1;
  const float pAk[4] = {pA0, pA1, pA2, pA3};

  const float* Br = ws + WS_BR;
  const float* Bi = ws + WS_BI;

  v8f accr = {};
  v8f acci = {};
#pragma unroll
  for (int kk = 0; kk < 4; ++kk) {
    const int k0 = 4 * kk + 2 * h;  // this lane's K pair
    v2f a;
    a.x = pAk[kk] * qx;             // amp[4kk + 2h]
    a.y = pAk[kk] * qy;             // amp[4kk + 2h + 1]
    v2f br, bi;
    br.x = Br[k0 * 16 + n];       br.y = Br[(k0 + 1) * 16 + n];
    bi.x = Bi[k0 * 16 + n];       bi.y = Bi[(k0 + 1) * 16 + n];
    accr = __builtin_amdgcn_wmma_f32_16x16x4_f32(false, a, false, br,
                                                 (short)0, accr, false, false);
    acci = __builtin_amdgcn_wmma_f32_16x16x4_f32(false, a, false, bi,
                                                 (short)0, acci, false, false);
  }

  // ---- probs -> PauliZ expectation partials -------------------------------
  // D-frag: lane holds (M = rr + 8h, N = n). The Z sign depends only on n, so
  // sum probs over the 8 rows in registers first, then one 16-lane butterfly.
  float probsum = 0.f;
#pragma unroll
  for (int rr = 0; rr < 8; ++rr) {
    const float re = accr[rr], im = acci[rr];
    probsum += re * re + im * im;
  }
  float v0 = ((n >> 3) & 1) ? -probsum : probsum;  // wire 0 <-> bit 3
  float v1 = ((n >> 2) & 1) ? -probsum : probsum;
  float v2 = ((n >> 1) & 1) ? -probsum : probsum;
  float v3 = ((n >> 0) & 1) ? -probsum : probsum;
#pragma unroll
  for (int st = 1; st <= 8; st <<= 1) {  // butterfly within each 16-lane half
    v0 += __shfl_xor(v0, st, 32);
    v1 += __shfl_xor(v1, st, 32);
    v2 += __shfl_xor(v2, st, 32);
    v3 += __shfl_xor(v3, st, 32);
  }
  if (n == 0) {  // lanes 0 (patches M=0..7) and 16 (M=8..15) of this tile
    const int slot = tile * 2 + h;
    eacc[slot][0] = v0; eacc[slot][1] = v1;
    eacc[slot][2] = v2; eacc[slot][3] = v3;
  }
  __syncthreads();

  // deterministic reduction over the 26 half-wave partials (invalid are 0)
  if (tid < 4) {
    float s = 0.f;
#pragma unroll
    for (int sl = 0; sl < NTILES * 2; ++sl) s += eacc[sl][tid];
    ws[WS_CTX + bb * 4 + tid] = s * (1.0f / (float)NPATCH);
  }
}

// ---------------------------------------------------------------------------
// Kernel C: logits = ctx @ W^T + b via one 16x16x4 WMMA, then log_softmax.
// One wave per 16 batch rows.
// ---------------------------------------------------------------------------
__global__ __launch_bounds__(32) void head_logsoftmax(
    const float* __restrict__ W, const float* __restrict__ bias,
    const float* __restrict__ ws, float* __restrict__ out) {
  const float* ctx = ws + WS_CTX;
  const int tile = blockIdx.x;
  const int lane = threadIdx.x;
  const int h = lane >> 4, n = lane & 15;

  const int rowA = tile * 16 + n;  // A-frag row
  v2f a;
  a.x = ctx[rowA * 4 + 2 * h];
  a.y = ctx[rowA * 4 + 2 * h + 1];

  v2f bf;  // B[k][n] = W[n][k], zero-padded for n >= 10
  bf.x = (n < 10) ? W[n * 4 + 2 * h]     : 0.f;
  bf.y = (n < 10) ? W[n * 4 + 2 * h + 1] : 0.f;

  v8f acc = {};
  acc = __builtin_amdgcn_wmma_f32_16x16x4_f32(false, a, false, bf,
                                              (short)0, acc, false, false);

  const float bv = (n < 10) ? bias[n] : 0.f;
#pragma unroll
  for (int rr = 0; rr < 8; ++rr) {
    const int row = tile * 16 + rr + 8 * h;
    const float logit = acc[rr] + bv;
    float lv = (n < 10) ? logit : -1.0e30f;
    float m = lv;
    m = fmaxf(m, __shfl_xor(m, 1, 32));
    m = fmaxf(m, __shfl_xor(m, 2, 32));
    m = fmaxf(m, __shfl_xor(m, 4, 32));
    m = fmaxf(m, __shfl_xor(m, 8, 32));
    float ex = (n < 10) ? __expf(logit - m) : 0.f;
    ex += __shfl_xor(ex, 1, 32);
    ex += __shfl_xor(ex, 2, 32);
    ex += __shfl_xor(ex, 4, 32);
    ex += __shfl_xor(ex, 8, 32);
    const float lse = __logf(ex) + m;
    if (n < 10) out[row * 10 + n] = logit - lse;
  }
}

// ---------------------------------------------------------------------------
extern "C" void kernel_launch(void* const* d_in, const int* in_sizes, int n_in,
                              void* d_out, int out_size, void* d_ws, size_t ws_size,
                              hipStream_t stream) {
  const float* x      = (const float*)d_in[0];  // (4096,1,28,28)
  const float* params = (const float*)d_in[1];  // (2,4,2)
  const float* W      = (const float*)d_in[2];  // (10,4)
  const float* b      = (const float*)d_in[3];  // (10,)
  float* ws  = (float*)d_ws;
  float* out = (float*)d_out;                   // (4096,10) log-probs

  build_unitary<<<1, 16, 0, stream>>>(params, ws);
  quanv_encode<<<BATCH, TPB_B, 0, stream>>>(x, ws);
  head_logsoftmax<<<BATCH / 16, 32, 0, stream>>>(W, b, ws, out);
}